// PCGradWarp_85057532330334
// MI455X (gfx1250) — compile-verified
//
#include <hip/hip_runtime.h>
#include <stdint.h>

// PCGrad via Gram-matrix reduction:
//   G = grads @ grads^T  (8x8, padded to 16x16, via V_WMMA_F32_16X16X4_F32)
//   scan in Gram space -> weights w[8]
//   out = sum_t w[t] * grads[t]
// Pass 1 uses CDNA5 async global->LDS DMA (ASYNCcnt) to double-buffer tiles
// with zero VGPR staging cost; two bandwidth-bound passes over the 256MB input.

typedef __attribute__((ext_vector_type(2))) float v2f;
typedef __attribute__((ext_vector_type(8))) float v8f;

#define T_TASKS   8
#define TILE_D    1024
#define LDS_PITCH (TILE_D + 4)   // +4 floats: bank-friendly WMMA read pattern

static __device__ __forceinline__ void async_wait0() {
#if __has_builtin(__builtin_amdgcn_s_wait_asynccnt)
    __builtin_amdgcn_s_wait_asynccnt(0);
#else
    asm volatile("s_wait_asynccnt 0x0" ::: "memory");
#endif
}

// ---------------------------------------------------------------- pass 1
__launch_bounds__(256)
__global__ void pcgrad_gram_wmma(const float* __restrict__ g,
                                 float* __restrict__ partial,
                                 int D, int ntiles)
{
    // physical rows 0..7  : buffer 0 (task rows)
    // physical rows 8..15 : buffer 1 (task rows)
    // physical rows 16..23: static zeros (padded WMMA rows 8..15)
    __shared__ float lds[24 * LDS_PITCH];

    const int tid  = threadIdx.x;          // 256 threads = 8 waves (wave32)
    const int lane = tid & 31;
    const int wave = tid >> 5;
    const int row  = lane & 15;            // WMMA M (A) == N (B) index
    const int hi   = lane >> 4;            // selects K pair {0,1} vs {2,3}

    // zero the shared padding rows once (visible after prologue barrier)
    for (int i = tid * 4; i < T_TASKS * LDS_PITCH; i += 1024)
        *(float4*)(lds + 16 * LDS_PITCH + i) = make_float4(0.f, 0.f, 0.f, 0.f);

    // low 32 bits of a flat shared-aperture address == LDS byte offset
    const unsigned ldsBase = (unsigned)(uintptr_t)(const void*)lds;

    // WMMA operand read pointers for each buffer; rows >= 8 hit the zero rows
    const int p0 = (row < T_TASKS) ? row           : (16 + row - T_TASKS);
    const int p1 = (row < T_TASKS) ? (T_TASKS+row) : (16 + row - T_TASKS);
    const float* lp0 = lds + p0 * LDS_PITCH + 2 * hi + wave * 128;
    const float* lp1 = lds + p1 * LDS_PITCH + 2 * hi + wave * 128;

    // issue 8 async b128 loads (one tile) into buffer `buf`; no dest VGPRs,
    // tracked by ASYNCcnt (CDNA5 GLOBAL_LOAD_ASYNC_TO_LDS_B128, GVS mode)
    auto issue_tile = [&](int tile, int buf) {
        const unsigned voff  = (unsigned)tile * (TILE_D * 4u) + (unsigned)tid * 16u;
        const unsigned lbase = ldsBase + (buf ? (T_TASKS * LDS_PITCH * 4u) : 0u)
                                       + (unsigned)tid * 16u;
        #pragma unroll
        for (int r = 0; r < T_TASKS; ++r) {
            const unsigned ldst = lbase + (unsigned)(r * LDS_PITCH * 4);
            const unsigned long long rowbase =
                (unsigned long long)(uintptr_t)(g + (size_t)r * D);
            asm volatile("global_load_async_to_lds_b128 %0, %1, %2"
                         :: "v"(ldst), "v"(voff), "s"(rowbase)
                         : "memory");
        }
    };

    v8f acc0 = {0.f,0.f,0.f,0.f,0.f,0.f,0.f,0.f};
    v8f acc1 = {0.f,0.f,0.f,0.f,0.f,0.f,0.f,0.f};

    int  tile = blockIdx.x;
    int  cur  = 0;
    bool have = (tile < ntiles);
    if (have) issue_tile(tile, 0);
    async_wait0();
    __syncthreads();

    while (have) {
        const int  next     = tile + gridDim.x;
        const bool haveNext = (next < ntiles);
        if (haveNext) issue_tile(next, cur ^ 1);   // DMA overlaps WMMAs below

        // wave w covers d-offsets [w*128, w*128+128) of the tile, K=4 per WMMA.
        // For G = A*A^T the CDNA5 A(16x4) and B(4x16) f32 per-lane layouts
        // coincide, so one ds_load_b64 serves as both operands. Two
        // accumulators break the WMMA->WMMA RAW chain.
        const float* lp = cur ? lp1 : lp0;
        #pragma unroll 8
        for (int j = 0; j < 32; j += 2) {
            v2f a0 = *(const v2f*)(lp + j * 4);
            v2f a1 = *(const v2f*)(lp + j * 4 + 4);
            acc0 = __builtin_amdgcn_wmma_f32_16x16x4_f32(
                false, a0, false, a0, (short)0, acc0, false, false);
            acc1 = __builtin_amdgcn_wmma_f32_16x16x4_f32(
                false, a1, false, a1, (short)0, acc1, false, false);
        }

        if (!haveNext) break;
        async_wait0();          // next buffer's DMA landed
        __syncthreads();        // everyone done reading current buffer
        cur ^= 1;
        tile = next;
    }

    __syncthreads();            // all WMMA reads done; safe to reuse LDS
    const v8f acc = acc0 + acc1;

    // C/D layout: VGPR v, lane l -> M = v + 8*(l>>4), N = l&15
    #pragma unroll
    for (int v = 0; v < 8; ++v) {
        const int M = v + 8 * hi;
        lds[wave * 256 + M * 16 + row] = acc[v];
    }
    __syncthreads();

    float s = 0.f;
    #pragma unroll
    for (int w = 0; w < 8; ++w) s += lds[w * 256 + tid];
    partial[(size_t)blockIdx.x * 256 + tid] = s;   // deterministic, no atomics
}

// ---------------------------------------------------------------- pass 2
__launch_bounds__(256)
__global__ void pcgrad_solve(const float* __restrict__ partial, int nb,
                             const float* __restrict__ g, int D, int tail_start,
                             float* __restrict__ wvec)
{
    __shared__ float G[256];
    const int tid = threadIdx.x;

    float s = 0.f;
    for (int b = 0; b < nb; ++b) s += partial[(size_t)b * 256 + tid];

    // scalar tail for D not divisible by TILE_D (no-op for the reference sizes)
    const int j = tid >> 4, k = tid & 15;
    if (j < T_TASKS && k < T_TASKS) {
        for (int d = tail_start; d < D; ++d)
            s += g[(size_t)j * D + d] * g[(size_t)k * D + d];
    }
    G[tid] = s;
    __syncthreads();

    if (tid == 0) {
        float w[8];
        #pragma unroll
        for (int i = 0; i < 8; ++i) w[i] = 0.f;
        for (int t = 0; t < 8; ++t) {
            float a[8];
            #pragma unroll
            for (int i = 0; i < 8; ++i) a[i] = (i == t) ? 1.f : 0.f;
            for (int kk = 0; kk < 8; ++kk) {
                float dot = 0.f;
                #pragma unroll
                for (int i = 0; i < 8; ++i) dot += a[i] * G[i * 16 + kk];
                const float coef = fminf(dot / G[kk * 16 + kk], 0.f);
                a[kk] -= coef;
            }
            #pragma unroll
            for (int i = 0; i < 8; ++i) w[i] += a[i];
        }
        #pragma unroll
        for (int i = 0; i < 8; ++i) wvec[i] = w[i];
    }
}

// ---------------------------------------------------------------- pass 3
__launch_bounds__(256)
__global__ void pcgrad_combine(const float* __restrict__ g,
                               const float* __restrict__ wvec,
                               float* __restrict__ out, int D)
{
    __shared__ float wsh[8];
    if (threadIdx.x < 8) wsh[threadIdx.x] = wvec[threadIdx.x];
    __syncthreads();
    float wr[8];
    #pragma unroll
    for (int t = 0; t < 8; ++t) wr[t] = wsh[t];

    const int n4 = D >> 2;
    for (int i = blockIdx.x * blockDim.x + threadIdx.x; i < n4;
         i += gridDim.x * blockDim.x) {
        float4 acc; acc.x = acc.y = acc.z = acc.w = 0.f;
        #pragma unroll
        for (int t = 0; t < 8; ++t) {
            const float4 v = *(const float4*)(g + (size_t)t * D + (size_t)i * 4);
            acc.x = fmaf(wr[t], v.x, acc.x);
            acc.y = fmaf(wr[t], v.y, acc.y);
            acc.z = fmaf(wr[t], v.z, acc.z);
            acc.w = fmaf(wr[t], v.w, acc.w);
        }
        *(float4*)(out + (size_t)i * 4) = acc;
    }
    // scalar tail (D % 4)
    for (int d = (n4 << 2) + blockIdx.x * blockDim.x + threadIdx.x; d < D;
         d += gridDim.x * blockDim.x) {
        float a = 0.f;
        #pragma unroll
        for (int t = 0; t < 8; ++t) a = fmaf(wr[t], g[(size_t)t * D + d], a);
        out[d] = a;
    }
}

// ---------------------------------------------------------------- launch
extern "C" void kernel_launch(void* const* d_in, const int* in_sizes, int n_in,
                              void* d_out, int out_size, void* d_ws, size_t ws_size,
                              hipStream_t stream)
{
    const float* g = (const float*)d_in[0];
    const int T = T_TASKS;
    const int D = in_sizes[0] / T;          // 8388608 for the reference setup
    float* out = (float*)d_out;
    float* ws  = (float*)d_ws;

    // partial Gram tiles: NB blocks * 256 floats; shrink if workspace is small
    int NB = 512;
    while (NB > 1 && ((size_t)(NB + 1) * 256 * sizeof(float)) > ws_size) NB >>= 1;

    const int ntiles     = D / TILE_D;
    const int tail_start = ntiles * TILE_D;

    float* partial = ws;
    float* wvec    = ws + (size_t)NB * 256;

    pcgrad_gram_wmma<<<NB, 256, 0, stream>>>(g, partial, D, ntiles);
    pcgrad_solve<<<1, 256, 0, stream>>>(partial, NB, g, D, tail_start, wvec);

    int cb = (D / 4 + 255) / 256;
    if (cb > 2048) cb = 2048;
    if (cb < 1) cb = 1;
    pcgrad_combine<<<cb, 256, 0, stream>>>(g, wvec, out, D);
}